// SpectralConv2D_T_52974126628968
// MI455X (gfx1250) — compile-verified
//
#include <hip/hip_runtime.h>

// Static problem configuration (matches the reference)
#define F_   64
#define K_   3
#define B_   64
#define H_   32
#define W_   32
#define C_   8
#define O1_  30
#define O2_  30
#define L_   (O1_*O2_)        // 900
#define N_   (H_*W_)          // 1024
#define M_   (B_*L_)          // 57600 rows of the im2col GEMM
#define KPAD 12               // K=9 padded to 3 wmma steps of 4

typedef float v2f __attribute__((ext_vector_type(2)));
typedef float v8f __attribute__((ext_vector_type(8)));

// --- prep 1: channel-mean image up[b*N + n] -------------------------------
__global__ void prep_up(const float* __restrict__ in, float* __restrict__ up) {
  int t = blockIdx.x * blockDim.x + threadIdx.x;      // 0 .. B*N-1
  if (t >= B_ * N_) return;
  const float* p = in + (size_t)t * C_;
  float s = 0.f;
#pragma unroll
  for (int c = 0; c < C_; ++c) s += p[c];
  up[t] = s * (1.0f / C_);
}

// --- prep 2: weights wsB[p*F + f] = omega[f,i,j]*(lin[i]-lout[j]), rows 9..11 = 0
__global__ void prep_base(const float* __restrict__ od, const float* __restrict__ otu,
                          const float* __restrict__ otl, const float* __restrict__ li,
                          const float* __restrict__ lo, float* __restrict__ wsB) {
  int t = blockIdx.x * blockDim.x + threadIdx.x;      // 0 .. KPAD*F-1
  if (t >= KPAD * F_) return;
  int f = t % F_;
  int p = t / F_;
  float v = 0.f;
  if (p < 9) {
    int i = p / 3, j = p % 3;
    float w = 0.f;
    if (i == j)       w = od[f * K_ + i];       // diag
    else if (p == 1)  w = otu[f * 3 + 0];       // strict upper (0,1)
    else if (p == 2)  w = otu[f * 3 + 1];       // (0,2)
    else if (p == 5)  w = otu[f * 3 + 2];       // (1,2)
    else if (p == 3)  w = otl[f * 3 + 0];       // strict lower (1,0)
    else if (p == 6)  w = otl[f * 3 + 1];       // (2,0)
    else              w = otl[f * 3 + 2];       // p==7 -> (2,1)
    v = w * (li[i] - lo[j]);
  }
  wsB[p * F_ + f] = v;
}

// --- main: im2col-on-the-fly GEMM via V_WMMA_F32_16X16X4_F32 ---------------
// One wave32 per 16x16 output tile; 3 wmma k-steps (K padded 9 -> 12).
__global__ void conv_wmma(const float* __restrict__ up, const float* __restrict__ wsB,
                          const float* __restrict__ enc, const float* __restrict__ dec,
                          float* __restrict__ out) {
  const int lane = threadIdx.x & 31;
  const int wave = blockIdx.x * (blockDim.x >> 5) + (threadIdx.x >> 5);
  const int nt   = wave & 3;          // N tile (F = 4*16)
  const int mt   = wave >> 2;         // M tile (0..3599)
  const int half = lane >> 4;         // 0: K=0,1   1: K=2,3 (per ISA A/B layout)
  const int r    = lane & 15;

  // This lane's im2col row (A) and output column (B/D)
  const int m     = mt * 16 + r;
  const int b     = m / L_;
  const int l     = m - b * L_;
  const int ro    = l / O2_;
  const int co    = l - ro * O2_;
  const int start = ro * W_ + co;
  const float dl  = dec[l];
  const float* upb = up + b * N_;
  const int fcol  = nt * 16 + r;

  v8f c = {};
#pragma unroll
  for (int s = 0; s < 3; ++s) {
    const int kb = 4 * s;
    const int p0 = kb + half * 2;     // lanes 0-15: kb,kb+1 ; lanes 16-31: kb+2,kb+3
    const int p1 = p0 + 1;

    v2f a, bb;
    {
      float x = 0.f;
      if (p0 < 9) {
        int di = p0 / 3, dj = p0 - di * 3;
        int n  = start + di * W_ + dj;
        x = upb[n] * (enc[n] - dl);
      }
      a.x = x;
    }
    {
      float x = 0.f;
      if (p1 < 9) {
        int di = p1 / 3, dj = p1 - di * 3;
        int n  = start + di * W_ + dj;
        x = upb[n] * (enc[n] - dl);
      }
      a.y = x;
    }
    bb.x = wsB[p0 * F_ + fcol];       // rows 9..11 are zero-filled, no guard
    bb.y = wsB[p1 * F_ + fcol];

    // D = A(16x4) * B(4x16) + C, f32 throughout
    c = __builtin_amdgcn_wmma_f32_16x16x4_f32(
        /*neg_a=*/false, a, /*neg_b=*/false, bb,
        /*c_mod=*/(short)0, c, /*reuse_a=*/false, /*reuse_b=*/false);
  }

  // C/D layout: VGPR g -> M = g (lanes 0-15) / 8+g (lanes 16-31); lane -> N
  const int row0 = mt * 16 + half * 8;
#pragma unroll
  for (int g = 0; g < 8; ++g) {
    out[(size_t)(row0 + g) * F_ + fcol] = fmaxf(c[g], 0.f);  // ReLU, (B,O1,O2,F) flat
  }
}

extern "C" void kernel_launch(void* const* d_in, const int* in_sizes, int n_in,
                              void* d_out, int out_size, void* d_ws, size_t ws_size,
                              hipStream_t stream) {
  const float* inp = (const float*)d_in[0];   // inputs      (B,H,W,C)
  const float* od  = (const float*)d_in[1];   // omega_diag  (F,K)
  const float* otu = (const float*)d_in[2];   // omega_triu  (1,F*3)
  const float* otl = (const float*)d_in[3];   // omega_tril  (1,F*3)
  const float* li  = (const float*)d_in[4];   // lambda_in   (K,1)
  const float* lo  = (const float*)d_in[5];   // lambda_out  (1,K)
  const float* enc = (const float*)d_in[6];   // use_encode  (1,N)
  const float* dec = (const float*)d_in[7];   // use_decode  (L,1)
  float* out = (float*)d_out;

  float* ws  = (float*)d_ws;
  float* up  = ws;                 // B*N floats (256 KB)
  float* wsB = ws + B_ * N_;       // KPAD*F floats

  prep_up  <<<(B_ * N_ + 255) / 256, 256, 0, stream>>>(inp, up);
  prep_base<<<(KPAD * F_ + 255) / 256, 256, 0, stream>>>(od, otu, otl, li, lo, wsB);

  // 3600 M-tiles * 4 N-tiles = 14400 waves; 8 waves (256 threads) per block
  conv_wmma<<<(M_ / 16) * 4 / 8, 256, 0, stream>>>(up, wsB, enc, dec, out);
}